// SparseBatchNorm1d_70970039599802
// MI455X (gfx1250) — compile-verified
//
#include <hip/hip_runtime.h>

// SparseBatchNorm over x[B=4, C=64, H=512, W=512] f32 (NCHW).
// Bandwidth-bound: ~0.77GB min traffic -> ~33us at 23.3TB/s.
// Stats reduction done on the CDNA5 matrix pipe: V_WMMA_F32_16X16X4_F32 with
// an all-ones A-matrix = exact f32 column-sum of the B operand.

#define CCH   64
#define HWC   262144           // H*W
#define BB    4
#define NPOS  (BB * HWC)       // 1,048,576 positions
#define EPSV  1e-5f

#define SPLITS 256             // position chunks per batch for the stats pass
#define CHUNK  (HWC / SPLITS)  // 1024 positions per block
#define ITERS  (CHUNK / 16)    // 16 positions per block-iteration (2 octets x 8)

typedef __attribute__((ext_vector_type(2))) float v2f;
typedef __attribute__((ext_vector_type(8))) float v8f;

// ws float layout: [0..63]=sum  [64..127]=sumsq  [128..191]=scale
//                  [192..255]=shift  [256]=count
__global__ void init_stats_kernel(float* __restrict__ ws) {
    int t = blockIdx.x * blockDim.x + threadIdx.x;
    if (t < 257) ws[t] = 0.0f;
}

// grid = BB*SPLITS blocks, 256 threads (8 waves).
// wave w: channel-group cg=w&3 (16 channels), octet oct=w>>2 (which 8 positions).
__global__ __launch_bounds__(256) void stats_kernel(const float* __restrict__ x,
                                                    float* __restrict__ ws) {
    __shared__ unsigned lds_pm[2];

    const int tid  = threadIdx.x;
    const int lane = tid & 31;
    const int wave = tid >> 5;
    const int cg   = wave & 3;       // channel group: channels [cg*16, cg*16+16)
    const int oct  = wave >> 2;      // position octet 0/1 within the 16-pos tile
    const int b    = blockIdx.x >> 8;          // / SPLITS
    const int sp   = blockIdx.x & (SPLITS - 1);
    const int n    = lane & 15;      // channel-within-group == WMMA column
    const int half = lane >> 4;      // which 4 positions of the octet

    const size_t row   = (size_t)(b * CCH + cg * 16 + n) * HWC;
    const int    sbase = sp * CHUNK + oct * 8 + half * 4;

    v2f ones; ones.x = 1.0f; ones.y = 1.0f;
    v8f acc_s = {};   // per-channel masked sum   (all 16 rows identical)
    v8f acc_q = {};   // per-channel masked sumsq
    unsigned cnt = 0;

    for (int it = 0; it < ITERS; ++it) {
        const float4 xv = *(const float4*)(x + row + (size_t)(sbase + it * 16));

        // per-position "any channel in my 16 > 0", packed into 8 bits (octet)
        unsigned pm = 0, bal;
        bal = __builtin_amdgcn_ballot_w32(xv.x > 0.0f);
        pm |= ((bal & 0xFFFFu) ? 1u : 0u) << 0;  pm |= ((bal >> 16) ? 1u : 0u) << 4;
        bal = __builtin_amdgcn_ballot_w32(xv.y > 0.0f);
        pm |= ((bal & 0xFFFFu) ? 1u : 0u) << 1;  pm |= ((bal >> 16) ? 1u : 0u) << 5;
        bal = __builtin_amdgcn_ballot_w32(xv.z > 0.0f);
        pm |= ((bal & 0xFFFFu) ? 1u : 0u) << 2;  pm |= ((bal >> 16) ? 1u : 0u) << 6;
        bal = __builtin_amdgcn_ballot_w32(xv.w > 0.0f);
        pm |= ((bal & 0xFFFFu) ? 1u : 0u) << 3;  pm |= ((bal >> 16) ? 1u : 0u) << 7;

        // OR the 4 channel-groups' partial masks per octet
        __syncthreads();
        if (tid < 2) lds_pm[tid] = 0u;
        __syncthreads();
        if (lane == 0) atomicOr(&lds_pm[oct], pm);
        __syncthreads();
        const unsigned comb = lds_pm[oct];
        cnt += (unsigned)__popc((int)comb);

        const int base = half * 4;
        float f0 = ((comb >> (base + 0)) & 1u) ? xv.x : 0.0f;
        float f1 = ((comb >> (base + 1)) & 1u) ? xv.y : 0.0f;
        float f2 = ((comb >> (base + 2)) & 1u) ? xv.z : 0.0f;
        float f3 = ((comb >> (base + 3)) & 1u) ? xv.w : 0.0f;

        v2f b1; b1.x = f0;      b1.y = f1;
        v2f b2; b2.x = f2;      b2.y = f3;
        v2f q1; q1.x = f0 * f0; q1.y = f1 * f1;
        v2f q2; q2.x = f2 * f2; q2.y = f3 * f3;

        // ones(16x4) x B(4x16): every D column n = sum over the 4 K-slots of
        // column n -> per-channel partial sum, accumulated in f32 on the matrix pipe.
        acc_s = __builtin_amdgcn_wmma_f32_16x16x4_f32(false, ones, false, b1,
                                                      (short)0, acc_s, false, false);
        acc_s = __builtin_amdgcn_wmma_f32_16x16x4_f32(false, ones, false, b2,
                                                      (short)0, acc_s, false, false);
        acc_q = __builtin_amdgcn_wmma_f32_16x16x4_f32(false, ones, false, q1,
                                                      (short)0, acc_q, false, false);
        acc_q = __builtin_amdgcn_wmma_f32_16x16x4_f32(false, ones, false, q2,
                                                      (short)0, acc_q, false, false);
    }

    // D VGPR0 lanes 0..15 hold row M=0 = per-channel sums for n = lane
    if (lane < 16) {
        atomicAdd(&ws[cg * 16 + lane],      acc_s[0]);
        atomicAdd(&ws[64 + cg * 16 + lane], acc_q[0]);
    }
    if (cg == 0 && lane == 0) atomicAdd(&ws[256], (float)cnt);
}

__global__ void finalize_kernel(const float* __restrict__ weight,
                                const float* __restrict__ bias,
                                float* __restrict__ ws) {
    int c = threadIdx.x;
    if (c >= CCH) return;
    float count = fmaxf(ws[256], 1.0f);
    float mean  = ws[c] / count;
    float var   = fmaxf(ws[64 + c] / count - mean * mean, 0.0f);
    float sc    = weight[c] * rsqrtf(var + EPSV);
    ws[128 + c] = sc;
    ws[192 + c] = bias[c] - mean * sc;
}

// One position per thread; channels cached in registers so the mask is free.
__global__ __launch_bounds__(256) void norm_kernel(const float* __restrict__ x,
                                                   const float* __restrict__ ws,
                                                   float* __restrict__ out) {
    __shared__ float s_scale[CCH];
    __shared__ float s_shift[CCH];
    if (threadIdx.x < CCH) {
        s_scale[threadIdx.x] = ws[128 + threadIdx.x];
        s_shift[threadIdx.x] = ws[192 + threadIdx.x];
    }
    __syncthreads();

    const int p = blockIdx.x * 256 + threadIdx.x;   // [0, NPOS)
    const int b = p >> 18;                          // / HWC
    const int s = p & (HWC - 1);
    const size_t base = (size_t)b * CCH * HWC + (size_t)s;

    float v[CCH];
    bool any = false;
#pragma unroll
    for (int c = 0; c < CCH; ++c) {
        v[c] = x[base + (size_t)c * HWC];
        any = any || (v[c] > 0.0f);
    }
#pragma unroll
    for (int c = 0; c < CCH; ++c) {
        float r = any ? (v[c] * s_scale[c] + s_shift[c]) : v[c];
        out[base + (size_t)c * HWC] = r;
    }
}

extern "C" void kernel_launch(void* const* d_in, const int* in_sizes, int n_in,
                              void* d_out, int out_size, void* d_ws, size_t ws_size,
                              hipStream_t stream) {
    const float* x      = (const float*)d_in[0];
    const float* weight = (const float*)d_in[1];
    const float* bias   = (const float*)d_in[2];
    float*       out    = (float*)d_out;
    float*       ws     = (float*)d_ws;

    init_stats_kernel<<<1, 512, 0, stream>>>(ws);
    stats_kernel<<<BB * SPLITS, 256, 0, stream>>>(x, ws);
    finalize_kernel<<<1, 64, 0, stream>>>(weight, bias, ws);
    norm_kernel<<<NPOS / 256, 256, 0, stream>>>(x, ws, out);
}